// SeparatedChannelChamferLoss_3977139716130
// MI455X (gfx1250) — compile-verified
//
#include <hip/hip_runtime.h>
#include <stdint.h>

// ---------------------------------------------------------------------------
// 1D separated-channel Chamfer loss, MI455X (gfx1250, wave32).
//   loss = sum_c [ mean_i min_j |pred[i,c]-target[j,c]|
//                + mean_j min_i |pred[i,c]-target[j,c]| ]
// Design: 6 (channel,direction) pairs x 32 query blocks; each block stages the
// 8192-element database channel into LDS via gfx1250 async global->LDS loads
// (stride-3 gather hidden behind ASYNCcnt), then every thread scans the LDS
// broadcast tile with v_sub/v_min|abs|. Block sum of the 256 per-thread minima
// is done on the matrix pipe: 4x V_WMMA_F32_16X16X4_F32 with A=ones so
// D[m,n] = colsum(B) accumulated over chunks; summing the 16 column sums gives
// the block total (arrangement-invariant). Partials -> tiny final reduce.
// ---------------------------------------------------------------------------

#define N_PTS   8192
#define TPB     256
#define QBLKS   (N_PTS / TPB)      // 32 query blocks per (channel,direction)
#define NPAIRS  6                  // 3 channels x 2 directions
#define NBLOCKS (NPAIRS * QBLKS)   // 192

typedef __attribute__((ext_vector_type(2))) float v2f;
typedef __attribute__((ext_vector_type(8))) float v8f;

__global__ __launch_bounds__(TPB) void chamfer1d_min_kernel(
    const float* __restrict__ pred,
    const float* __restrict__ target,
    float* __restrict__ partials)
{
    __shared__ __align__(16) float sdb[N_PTS];  // database channel slice, 32 KB
    __shared__ float sred[TPB];                 // per-thread minima

    const int tid  = threadIdx.x;
    const int pair = blockIdx.x / QBLKS;        // 0..5
    const int qblk = blockIdx.x % QBLKS;        // 0..31
    const int dir  = pair / 3;                  // 0: pred->target, 1: target->pred
    const int ch   = pair % 3;                  // x / y / z

    const float* qbase = dir ? target : pred;
    const float* dbase = dir ? pred   : target;

    // --- stage database channel into LDS with gfx1250 async loads -----------
    // GVS addressing: 64-bit SGPR base + per-lane 32-bit byte offset; LDS
    // destination byte address carried in a VGPR (low 32 bits of flat addr).
    #pragma unroll
    for (int k = 0; k < N_PTS / TPB; ++k) {     // 32 x b32 per thread
        const int j = tid + k * TPB;
        uint32_t lds_addr = (uint32_t)(uintptr_t)(&sdb[j]);
        uint32_t goff     = (uint32_t)((3u * (uint32_t)j + (uint32_t)ch) * 4u);
        asm volatile("global_load_async_to_lds_b32 %0, %1, %2"
                     :
                     : "v"(lds_addr), "v"(goff), "s"(dbase)
                     : "memory");
    }

    // overlap: fetch this thread's query value while the DMA is in flight
    const int   q  = qblk * TPB + tid;
    const float qv = qbase[3 * q + ch];

    asm volatile("s_wait_asynccnt 0" ::: "memory");
    __syncthreads();

    // --- broadcast scan: min_j |qv - db[j]| ---------------------------------
    float m = 3.402823466e38f;
    const float4* s4 = reinterpret_cast<const float4*>(sdb);
    #pragma unroll 4
    for (int k = 0; k < N_PTS / 4; ++k) {
        float4 v = s4[k];                       // ds_load_b128, lane-broadcast
        m = fminf(m, fabsf(qv - v.x));
        m = fminf(m, fabsf(qv - v.y));
        m = fminf(m, fabsf(qv - v.z));
        m = fminf(m, fabsf(qv - v.w));
    }

    sred[tid] = m;
    __syncthreads();

    // --- block sum of 256 minima on the matrix pipe -------------------------
    // A = ones(16x4)  =>  D[m,n] = sum_k B[k,n]; accumulate 4 chunks of 64
    // values through C, then add the 16 column sums. Total is invariant to the
    // B VGPR striping, so the load pattern below is safe. Executed uniformly
    // by all waves (EXEC must be all 1s for WMMA; result identical per wave).
    const int lane = tid & 31;
    v2f a; a[0] = 1.0f; a[1] = 1.0f;
    v8f acc = {};
    #pragma unroll
    for (int t = 0; t < 4; ++t) {
        v2f b;
        b[0] = sred[t * 64 + lane];
        b[1] = sred[t * 64 + 32 + lane];
        acc = __builtin_amdgcn_wmma_f32_16x16x4_f32(
            /*neg_a=*/false, a, /*neg_b=*/false, b,
            /*c_mod=*/(short)0, acc, /*reuse_a=*/false, /*reuse_b=*/false);
    }
    // lanes 0-15 of acc[0] hold D[0][0..15] (every row replicates the colsums)
    float col = acc[0];
    #pragma unroll
    for (int off = 8; off >= 1; off >>= 1)
        col += __shfl_xor(col, off, 32);

    if (tid == 0)
        partials[blockIdx.x] = col;             // sum of this block's 256 minima
}

__global__ __launch_bounds__(TPB) void chamfer1d_reduce_kernel(
    const float* __restrict__ partials,
    float* __restrict__ out)
{
    __shared__ float s[TPB];
    const int tid = threadIdx.x;
    s[tid] = (tid < NBLOCKS) ? partials[tid] : 0.0f;
    __syncthreads();
    #pragma unroll
    for (int off = TPB / 2; off > 0; off >>= 1) {
        if (tid < off) s[tid] += s[tid + off];
        __syncthreads();
    }
    if (tid == 0)
        out[0] = s[0] * (1.0f / (float)N_PTS);  // converts the 6 sums to means
}

extern "C" void kernel_launch(void* const* d_in, const int* in_sizes, int n_in,
                              void* d_out, int out_size, void* d_ws, size_t ws_size,
                              hipStream_t stream)
{
    const float* pred   = (const float*)d_in[0];   // [8192,3] f32
    const float* target = (const float*)d_in[1];   // [8192,3] f32
    float* out      = (float*)d_out;               // scalar f32
    float* partials = (float*)d_ws;                // 192 floats of scratch

    chamfer1d_min_kernel<<<NBLOCKS, TPB, 0, stream>>>(pred, target, partials);
    chamfer1d_reduce_kernel<<<1, TPB, 0, stream>>>(partials, out);
}